// IAMC_30906584662137
// MI455X (gfx1250) — compile-verified
//
#include <hip/hip_runtime.h>
#include <hip/hip_bf16.h>
#include <climits>

// ---------------------------------------------------------------------------
// IAMC loss: cosine-sim of every row vs anchor row, masked means, over 6
// modalities. Memory-bound (402 MB @ 23.3 TB/s ~ 17us). Dot products go
// through V_WMMA_F32_16X16X4_F32; squared norms via VALU; anchor row staged
// in LDS; deterministic fixed-order reductions (no float atomics).
// A-operands are loaded directly in WMMA lane layout (global_load_b64 per
// lane) so the inner loop has no cross-lane redistribution.
// ---------------------------------------------------------------------------

typedef float v2f __attribute__((ext_vector_type(2)));
typedef float v8f __attribute__((ext_vector_type(8)));

#define NROW 8192
#define NCOL 2048
#define NMOD 6
#define TILE 16
#define WAVES 8          // 256 threads / 32 lanes
#define KCHUNK 256       // columns per wave (2048 / 8 waves)

// ---------------- Phase 0: anchor = first index with label == 0 -------------
__global__ __launch_bounds__(1024) void iamc_find_anchor(
    const int* __restrict__ label, int n, int* __restrict__ out) {
    __shared__ int smin[1024];
    int t = threadIdx.x;
    int best = INT_MAX;
    for (int i = t; i < n; i += blockDim.x)
        if (label[i] == 0) best = min(best, i);
    smin[t] = best;
    __syncthreads();
    for (int s = 512; s > 0; s >>= 1) {
        if (t < s) smin[t] = min(smin[t], smin[t + s]);
        __syncthreads();
    }
    if (t == 0) out[0] = smin[0];
}

// ---------------- Phase 1: per-row dot-with-anchor and squared norm ---------
// One block = one (modality, 16-row tile). 8 waves split the K dimension.
__global__ __launch_bounds__(256) void iamc_phase1(
    const float* __restrict__ x0, const float* __restrict__ x1,
    const float* __restrict__ x2, const float* __restrict__ x3,
    const float* __restrict__ x4, const float* __restrict__ x5,
    const int* __restrict__ anchorP,
    float* __restrict__ gDot, float* __restrict__ gSq) {

    __shared__ float sAnc[NCOL];          // 8 KB anchor row
    __shared__ float sDot[WAVES][TILE];
    __shared__ float sSq[WAVES][TILE];

    const int m    = blockIdx.x >> 9;     // / 512 tiles per modality
    const int tile = blockIdx.x & 511;
    const int row0 = tile * TILE;

    const float* X = x0;
    if (m == 1) X = x1;
    else if (m == 2) X = x2;
    else if (m == 3) X = x3;
    else if (m == 4) X = x4;
    else if (m == 5) X = x5;

    const int anchor = *anchorP;

    // Stage anchor row of this modality into LDS (float4 cooperative copy).
    const float4* ancRow = (const float4*)(X + (size_t)anchor * NCOL);
    for (int i = threadIdx.x; i < NCOL / 4; i += 256)
        ((float4*)sAnc)[i] = ancRow[i];
    __syncthreads();

    const int lane = threadIdx.x & 31;
    const int wave = threadIdx.x >> 5;
    const int r    = lane & 15;
    const bool hi  = lane >= 16;
    const int koff = hi ? 2 : 0;          // WMMA A layout: lo lanes K0,K1; hi K2,K3

    const float* rowPtr = X + (size_t)(row0 + r) * NCOL;

    v8f c = {0.f, 0.f, 0.f, 0.f, 0.f, 0.f, 0.f, 0.f};
    float sq = 0.f;

    const int kbeg = wave * KCHUNK;
    const int kend = kbeg + KCHUNK;
    #pragma unroll 4
    for (int kb = kbeg; kb < kend; kb += 8) {
        // Each lane loads exactly its WMMA A slots: 2 x global_load_b64.
        const float2 f0 = *(const float2*)(rowPtr + kb + koff);       // k = kb..kb+3
        const float2 f1 = *(const float2*)(rowPtr + kb + 4 + koff);   // k = kb+4..kb+7

        sq = fmaf(f0.x, f0.x, sq);
        sq = fmaf(f0.y, f0.y, sq);
        sq = fmaf(f1.x, f1.x, sq);
        sq = fmaf(f1.y, f1.y, sq);

        // B: anchor chunk, constant across all 16 N columns, K striped to
        // match A. ds_load_2addr_b64 pairs both chunks.
        const float2 bb0 = *(const float2*)&sAnc[kb + koff];
        const float2 bb1 = *(const float2*)&sAnc[kb + 4 + koff];

        v2f a0; a0.x = f0.x;  a0.y = f0.y;
        v2f a1; a1.x = f1.x;  a1.y = f1.y;
        v2f b0; b0.x = bb0.x; b0.y = bb0.y;
        v2f b1; b1.x = bb1.x; b1.y = bb1.y;

        c = __builtin_amdgcn_wmma_f32_16x16x4_f32(false, a0, false, b0,
                                                  (short)0, c, false, false);
        c = __builtin_amdgcn_wmma_f32_16x16x4_f32(false, a1, false, b1,
                                                  (short)0, c, false, false);
    }

    // Squared-norm partial: lane l and l+16 both cover row r -> pair-sum.
    sq += __shfl_xor(sq, 16, 32);
    if (lane < 16) sSq[wave][lane] = sq;

    // Dots: every column of D is identical. Lane 0 holds rows 0-7 in c[0..7],
    // lane 16 holds rows 8-15 (C layout: VGPR v, lanes 0-15 -> M=v, 16-31 -> M=v+8).
    if (lane == 0) {
        sDot[wave][0] = c[0]; sDot[wave][1] = c[1];
        sDot[wave][2] = c[2]; sDot[wave][3] = c[3];
        sDot[wave][4] = c[4]; sDot[wave][5] = c[5];
        sDot[wave][6] = c[6]; sDot[wave][7] = c[7];
    }
    if (lane == 16) {
        sDot[wave][8]  = c[0]; sDot[wave][9]  = c[1];
        sDot[wave][10] = c[2]; sDot[wave][11] = c[3];
        sDot[wave][12] = c[4]; sDot[wave][13] = c[5];
        sDot[wave][14] = c[6]; sDot[wave][15] = c[7];
    }
    __syncthreads();

    // Fixed-order cross-wave reduction -> global (deterministic).
    if (wave == 0 && lane < 16) {
        float d = 0.f, s = 0.f;
        #pragma unroll
        for (int w = 0; w < WAVES; ++w) { d += sDot[w][lane]; s += sSq[w][lane]; }
        const size_t o = (size_t)m * NROW + row0 + lane;
        gDot[o] = d;
        gSq[o]  = s;
    }
}

// ---------------- Phase 2: masked means and final scalar --------------------
__global__ __launch_bounds__(256) void iamc_finalize(
    const float* __restrict__ gDot, const float* __restrict__ gSq,
    const int* __restrict__ label, const int* __restrict__ anchorP,
    float* __restrict__ out) {

    __shared__ float sP[256], sN[256];
    __shared__ int   cP[256], cN[256];

    const int anchor = *anchorP;
    const int t = threadIdx.x;
    float total = 0.f;

    for (int m = 0; m < NMOD; ++m) {
        const float normA = sqrtf(gSq[(size_t)m * NROW + anchor]);
        float ps = 0.f, ns = 0.f;
        int pc = 0, nc = 0;
        for (int i = t; i < NROW; i += 256) {
            const size_t o = (size_t)m * NROW + i;
            const float sim = gDot[o] / (sqrtf(gSq[o]) * normA);
            const int lb = label[i];
            if (lb == 0) {
                if (i != anchor) { ps += sim; pc++; }
            } else {
                ns += sim; nc++;
            }
        }
        sP[t] = ps; sN[t] = ns; cP[t] = pc; cN[t] = nc;
        __syncthreads();
        for (int s = 128; s > 0; s >>= 1) {
            if (t < s) {
                sP[t] += sP[t + s]; sN[t] += sN[t + s];
                cP[t] += cP[t + s]; cN[t] += cN[t + s];
            }
            __syncthreads();
        }
        if (t == 0) {
            const float pm = (cP[0] > 0) ? sP[0] / (float)cP[0] : 0.f;
            const float nm = (cN[0] > 0) ? sN[0] / (float)cN[0] : 0.f;
            total += 2.0f - pm + nm;
        }
        __syncthreads();   // shared arrays reused next m
    }
    if (t == 0) out[0] = total / (float)NMOD;
}

// ---------------------------------------------------------------------------
extern "C" void kernel_launch(void* const* d_in, const int* in_sizes, int n_in,
                              void* d_out, int out_size, void* d_ws, size_t ws_size,
                              hipStream_t stream) {
    (void)in_sizes; (void)n_in; (void)out_size; (void)ws_size;

    const float* X0 = (const float*)d_in[0];   // p_t
    const float* X1 = (const float*)d_in[1];   // p_v
    const float* X2 = (const float*)d_in[2];   // p_a
    const float* X3 = (const float*)d_in[3];   // s_t
    const float* X4 = (const float*)d_in[4];   // s_v
    const float* X5 = (const float*)d_in[5];   // s_a
    const int*  label = (const int*)d_in[6];

    int*   anchorP = (int*)d_ws;
    float* gDot    = ((float*)d_ws) + 16;            // 64B-aligned region
    float* gSq     = gDot + (size_t)NMOD * NROW;

    iamc_find_anchor<<<1, 1024, 0, stream>>>(label, NROW, anchorP);

    const int blocks = NMOD * (NROW / TILE);         // 6 * 512 = 3072
    iamc_phase1<<<blocks, 256, 0, stream>>>(X0, X1, X2, X3, X4, X5,
                                            anchorP, gDot, gSq);

    iamc_finalize<<<1, 256, 0, stream>>>(gDot, gSq, label, anchorP,
                                         (float*)d_out);
}